// SAGEBipartiteLayer_35648228556938
// MI455X (gfx1250) — compile-verified
//
#include <hip/hip_runtime.h>
#include <hip/hip_bf16.h>

#define DIM 128

typedef float v2f __attribute__((ext_vector_type(2)));
typedef float v8f __attribute__((ext_vector_type(8)));

// ---------------- zero the accumulation / output buffer ----------------
__global__ void zero_f32(float* __restrict__ p, int n) {
  int i = blockIdx.x * blockDim.x + threadIdx.x;
  int stride = gridDim.x * blockDim.x;
  for (; i < n; i += stride) p[i] = 0.0f;
}

// ---------------- SpMM scatter: dst[row] += val * src[col] ----------------
// One wave32 per edge; each lane handles 4 consecutive floats of the 128-wide
// row (float4 load, 4 hardware f32 atomics, no return -> global_atomic_add_f32).
__global__ void __launch_bounds__(256)
spmm_scatter(const int* __restrict__ rows, const int* __restrict__ cols,
             const float* __restrict__ vals, const float* __restrict__ src,
             float* __restrict__ dst, int nedges) {
  const int wid  = (blockIdx.x * blockDim.x + threadIdx.x) >> 5;
  const int lane = threadIdx.x & 31;
  if (wid >= nedges) return;
  const int   r = rows[wid];
  const int   c = cols[wid];
  const float v = vals[wid];
  const float4 x = *(const float4*)(src + (size_t)c * DIM + lane * 4);
  float* d = dst + (size_t)r * DIM + lane * 4;
  unsafeAtomicAdd(d + 0, x.x * v);
  unsafeAtomicAdd(d + 1, x.y * v);
  unsafeAtomicAdd(d + 2, x.z * v);
  unsafeAtomicAdd(d + 3, x.w * v);
}

// ---------------- fused GEMM pass: acc += Xtile @ W^T (WMMA f32 16x16x4) ----
// A (16x4 f32): lanes 0-15 hold M=lane&15, K = 4*kc + {0,1};
//               lanes 16-31 hold K = 4*kc + {2,3}.   (ISA 7.12.2)
// B (4x16 f32): B[k][n] = W[n][k]; same per-lane K offsets, lane&15 = n.
__device__ __forceinline__ void mma_pass(const float* __restrict__ X,
                                         const float* __restrict__ W,
                                         int m0, int nrows, int l, int h,
                                         v8f acc[8]) {
  int mr = m0 + l;
  if (mr > nrows - 1) mr = nrows - 1;   // clamp tail rows (loads only)
  const float* xrow = X + (size_t)mr * DIM;
#pragma unroll 2
  for (int kc = 0; kc < DIM / 4; ++kc) {
    const int k0 = kc * 4 + 2 * h;
    v2f a;
    a.x = xrow[k0];
    a.y = xrow[k0 + 1];
#pragma unroll
    for (int t = 0; t < 8; ++t) {
      const float* wrow = W + (size_t)(16 * t + l) * DIM;
      v2f b;
      b.x = wrow[k0];
      b.y = wrow[k0 + 1];
      acc[t] = __builtin_amdgcn_wmma_f32_16x16x4_f32(
          false, a, false, b, (short)0, acc[t], false, false);
    }
  }
}

// out = relu(Xself @ Wself^T + out(=neigh) @ Wneigh^T), computed in place.
// 8 waves per block; each wave owns a 16-row strip x all 128 output columns,
// so it fully consumes its own neigh rows before overwriting them.
__global__ void __launch_bounds__(256)
sage_gemm_relu(const float* __restrict__ Xself,
               const float* __restrict__ Wself,
               const float* __restrict__ Wneigh,
               float* __restrict__ out, int nrows) {
  const int lane  = threadIdx.x & 31;
  const int wave  = threadIdx.x >> 5;
  const int strip = blockIdx.x * 8 + wave;
  const int nstrips = (nrows + 15) / 16;
  if (strip >= nstrips) return;   // whole-wave exit: EXEC stays all-1 for WMMA
  const int m0 = strip * 16;
  const int l = lane & 15;
  const int h = lane >> 4;

  v8f acc[8];
#pragma unroll
  for (int t = 0; t < 8; ++t) acc[t] = {};

  mma_pass(Xself, Wself,  m0, nrows, l, h, acc);   // self term
  mma_pass(out,   Wneigh, m0, nrows, l, h, acc);   // neigh term (in-place read)

  // C/D layout: VGPR r -> row m0+r (lanes 0-15) / m0+r+8 (lanes 16-31), N=l.
#pragma unroll
  for (int t = 0; t < 8; ++t) {
#pragma unroll
    for (int r = 0; r < 8; ++r) {
      const int m = m0 + r + 8 * h;
      if (m < nrows) {
        const float vv = acc[t][r];
        out[(size_t)m * DIM + 16 * t + l] = vv > 0.0f ? vv : 0.0f;
      }
    }
  }
}

extern "C" void kernel_launch(void* const* d_in, const int* in_sizes, int n_in,
                              void* d_out, int out_size, void* d_ws, size_t ws_size,
                              hipStream_t stream) {
  (void)n_in; (void)d_ws; (void)ws_size;
  const float* Xg_self  = (const float*)d_in[0];
  const float* Xs_self  = (const float*)d_in[1];
  const float* Xs_for_g = (const float*)d_in[2];
  const float* Xg_for_s = (const float*)d_in[3];
  const int*   row_gs   = (const int*)d_in[4];
  const int*   col_gs   = (const int*)d_in[5];
  const float* val_gs   = (const float*)d_in[6];
  const int*   row_sg   = (const int*)d_in[7];
  const int*   col_sg   = (const int*)d_in[8];
  const float* val_sg   = (const float*)d_in[9];
  const float* Wg_self  = (const float*)d_in[10];
  const float* Wg_neigh = (const float*)d_in[11];
  const float* Ws_self  = (const float*)d_in[12];
  const float* Ws_neigh = (const float*)d_in[13];

  const int E  = in_sizes[4];
  const int NG = in_sizes[0] / DIM;
  const int NS = in_sizes[1] / DIM;

  float* out_g = (float*)d_out;
  float* out_s = out_g + (size_t)NG * DIM;

  // 1) zero d_out (it doubles as the neigh accumulator; d_out is poisoned).
  zero_f32<<<2048, 256, 0, stream>>>(out_g, out_size);

  // 2) scatter-add SpMM into d_out (one wave per edge).
  const int sblocks = (E * 32 + 255) / 256;
  spmm_scatter<<<sblocks, 256, 0, stream>>>(row_gs, col_gs, val_gs,
                                            Xs_for_g, out_g, E);
  spmm_scatter<<<sblocks, 256, 0, stream>>>(row_sg, col_sg, val_sg,
                                            Xg_for_s, out_s, E);

  // 3) fused dense GEMM + ReLU, in place over the neigh accumulators.
  const int gblocks = ((NG + 15) / 16 + 7) / 8;
  sage_gemm_relu<<<gblocks, 256, 0, stream>>>(Xg_self, Wg_self, Wg_neigh,
                                              out_g, NG);
  const int sblocks2 = ((NS + 15) / 16 + 7) / 8;
  sage_gemm_relu<<<sblocks2, 256, 0, stream>>>(Xs_self, Ws_self, Ws_neigh,
                                               out_s, NS);
}